// GNNAtomicLast_20366734917671
// MI455X (gfx1250) — compile-verified
//
#include <hip/hip_runtime.h>

typedef __attribute__((ext_vector_type(2))) float v2f;
typedef __attribute__((ext_vector_type(8))) float v8f;

#define BS 8
#define NN 1024        // nodes
#define JJ 4
#define FF 32
#define NC 2
#define KD (NN * JJ)   // 4096 reduction length per row

#define SEG   8                      // K segments (over m)
#define MSEG  (NN / SEG)             // 128 m-steps per segment
#define WAVES 4
#define TILES 2                      // 16-row tiles per wave
#define ROWS_PER_BLOCK (WAVES * TILES * 16)   // 128
#define RBLK  (NN / ROWS_PER_BLOCK)           // 8 row-blocks per batch

// ---------------------------------------------------------------------------
// Seed output with the bias (d_out is poisoned by the harness; atomics add on
// top of this).
// ---------------------------------------------------------------------------
__global__ void init_out_kernel(float* __restrict__ out, const float* __restrict__ fc_b) {
    int i = blockIdx.x * blockDim.x + threadIdx.x;
    if (i < BS * NN * NC) out[i] = fc_b[i & 1];   // NC == 2, c is fastest index
}

// ---------------------------------------------------------------------------
// Main fused kernel: out[b,n,c] += sum_{m,j} WW[b,n,m,j] * y[b,m,j,c]
// with y computed on the fly into LDS in the WMMA B-fragment layout.
//
// WMMA V_WMMA_F32_16X16X4_F32 per m-step:
//   A[16x4]  = WW[b, n0:n0+16, m, 0:4]   (j contiguous => per-lane b64 load)
//   B[4x16]  = y[b, m, j, c] in cols 0..1, columns 2..15 are don't-care
//   C[16x16] accumulates over m; only columns 0,1 stored.
// ---------------------------------------------------------------------------
__global__ __launch_bounds__(WAVES * 32) void gmul_fc_kernel(
        const float* __restrict__ WW,
        const float* __restrict__ x,
        const float* __restrict__ fc_w,
        float* __restrict__ out) {
    // B-fragment streams: Z[cls][mp], cls = (lane>>4)*2 + (lane&1)
    //   cls -> (half = cls>>1, c = cls&1):  Z = { y[4m+2h][c], y[4m+2h+1][c] }
    __shared__ v2f Z[4 * MSEG];   // 4 KB

    const int b    = blockIdx.z;
    const int seg  = blockIdx.y;
    const int rblk = blockIdx.x;
    const int tid  = threadIdx.x;
    const int mStart = seg * MSEG;

    // ---- build the y slice for this K segment into LDS --------------------
    for (int idx = tid; idx < 4 * MSEG; idx += WAVES * 32) {
        const int cls = idx / MSEG;
        const int mp  = idx & (MSEG - 1);
        const int h   = cls >> 1;
        const int c   = cls & 1;
        const int m   = mStart + mp;
        const float* xr = x + ((size_t)b * NN + m) * FF;
        v2f val;
        #pragma unroll
        for (int e = 0; e < 2; ++e) {
            const int j = 2 * h + e;
            const float* wr = fc_w + c * (JJ * FF) + j * FF;
            float s = 0.f;
            #pragma unroll
            for (int f = 0; f < FF; ++f) s += xr[f] * wr[f];
            val[e] = s;
        }
        Z[idx] = val;
    }
    __syncthreads();

    // ---- WMMA accumulation over the segment --------------------------------
    const int lane = tid & 31;
    const int wave = tid >> 5;
    const int half = lane >> 4;        // 0: K=0,1   1: K=2,3
    const int colc = lane & 15;        // output column (class) this lane holds
    const int cls  = half * 2 + (lane & 1);
    const int n0   = rblk * ROWS_PER_BLOCK + wave * (TILES * 16);

    // per-lane A base: row n0+(lane&15), dword offset 4*m + 2*half
    const float* a0 = WW + ((size_t)b * NN + n0 + colc) * (size_t)KD
                         + (size_t)mStart * 4 + half * 2;
    const float* a1 = a0 + (size_t)16 * KD;       // second 16-row tile
    const v2f* zb = &Z[cls * MSEG];

    v8f acc0 = {};
    v8f acc1 = {};

    #pragma unroll 4
    for (int mp = 0; mp < MSEG; ++mp) {
        v2f bfrag = zb[mp];
        v2f af0 = *(const v2f*)(a0 + (size_t)mp * 4);
        v2f af1 = *(const v2f*)(a1 + (size_t)mp * 4);
        acc0 = __builtin_amdgcn_wmma_f32_16x16x4_f32(
                   false, af0, false, bfrag, (short)0, acc0, false, false);
        acc1 = __builtin_amdgcn_wmma_f32_16x16x4_f32(
                   false, af1, false, bfrag, (short)0, acc1, false, false);
    }

    // ---- store: C layout lane holds N=colc, VGPR r -> M = half*8 + r -------
    if (colc < NC) {
        #pragma unroll
        for (int r = 0; r < 8; ++r) {
            const int row0 = n0 + half * 8 + r;
            atomicAdd(&out[((size_t)b * NN + row0) * NC + colc], acc0[r]);
            atomicAdd(&out[((size_t)b * NN + row0 + 16) * NC + colc], acc1[r]);
        }
    }
}

// ---------------------------------------------------------------------------
extern "C" void kernel_launch(void* const* d_in, const int* in_sizes, int n_in,
                              void* d_out, int out_size, void* d_ws, size_t ws_size,
                              hipStream_t stream) {
    const float* WW   = (const float*)d_in[0];   // (8,1024,1024,4) f32
    const float* x    = (const float*)d_in[1];   // (8,1024,32)     f32
    const float* fc_w = (const float*)d_in[2];   // (2,128)         f32
    const float* fc_b = (const float*)d_in[3];   // (2,)            f32
    float* out = (float*)d_out;                  // (8,1024,2)      f32
    (void)in_sizes; (void)n_in; (void)out_size; (void)d_ws; (void)ws_size;

    init_out_kernel<<<(BS * NN * NC + 255) / 256, 256, 0, stream>>>(out, fc_b);

    dim3 grid(RBLK, SEG, BS);                    // 8 x 8 x 8 = 512 blocks
    gmul_fc_kernel<<<grid, WAVES * 32, 0, stream>>>(WW, x, fc_w, out);
}